// GCNLayer_25907242729954
// MI455X (gfx1250) — compile-verified
//
#include <hip/hip_runtime.h>
#include <hip/hip_bf16.h>

typedef __attribute__((ext_vector_type(2))) float v2f;
typedef __attribute__((ext_vector_type(8))) float v8f;

#define IN_SIZE 64
#define OUT_SIZE 64

// out[n, c] = sum_r bias[r, c], vectorized as float4
__global__ __launch_bounds__(256) void rgcn_init_out(
    const float* __restrict__ bias, float* __restrict__ out, int n4, int n_rel) {
  int i = blockIdx.x * blockDim.x + threadIdx.x;
  if (i >= n4) return;
  int c = (i * 4) & (OUT_SIZE - 1);
  float4 b = make_float4(0.f, 0.f, 0.f, 0.f);
  for (int r = 0; r < n_rel; ++r) {
    const float* br = bias + r * OUT_SIZE + c;
    b.x += br[0]; b.y += br[1]; b.z += br[2]; b.w += br[3];
  }
  ((float4*)out)[i] = b;
}

__global__ __launch_bounds__(256) void rgcn_zero(float4* __restrict__ p, int n4) {
  int i = blockIdx.x * blockDim.x + threadIdx.x;
  if (i < n4) p[i] = make_float4(0.f, 0.f, 0.f, 0.f);
}

// One edge per wave32. Lane l handles columns 2l and 2l+1:
// coalesced 256B row gather + coalesced fp32 atomics into agg[dst].
__global__ __launch_bounds__(256) void rgcn_scatter(
    const float* __restrict__ inp, const int* __restrict__ src,
    const int* __restrict__ dst, const float* __restrict__ eval,
    float* __restrict__ agg, int n_edges) {
  const int wid = (blockIdx.x * 256 + threadIdx.x) >> 5;
  if (wid >= n_edges) return;
  const int lane = threadIdx.x & 31;
  const int s = src[wid];
  const int d = dst[wid];
  const float v = eval[wid];
  const int col = lane * 2;
  const float2 m = *(const float2*)(inp + (size_t)s * IN_SIZE + col);
  float* p = agg + (size_t)d * IN_SIZE + col;
  unsafeAtomicAdd(p, m.x * v);      // global_atomic_add_f32
  unsafeAtomicAdd(p + 1, m.y * v);
}

// out(100000x64) += agg(100000x64) @ W(64x64), f32 WMMA 16x16x4.
// One 16x16 output tile per wave; 8 waves/block; W staged in LDS.
__global__ __launch_bounds__(256) void rgcn_gemm_wmma(
    const float* __restrict__ agg, const float* __restrict__ W,
    float* __restrict__ out, int num_tiles) {
  __shared__ float sW[IN_SIZE * OUT_SIZE];  // 16 KB
  const int t = threadIdx.x;
  {
    const float4* Wv = (const float4*)W;
    float4* sWv = (float4*)sW;
#pragma unroll
    for (int i = 0; i < (IN_SIZE * OUT_SIZE / 4) / 256; ++i)
      sWv[t + i * 256] = Wv[t + i * 256];
  }
  __syncthreads();

  const int wave = t >> 5;
  const int lane = t & 31;
  const int half = lane >> 4;  // 0 -> lanes 0..15, 1 -> lanes 16..31
  const int l16 = lane & 15;

  const int tile = blockIdx.x * 8 + wave;  // uniform per wave
  if (tile >= num_tiles) return;
  const int rowBase = (tile >> 2) * 16;
  const int colBase = (tile & 3) * 16;

  // Load C tile from out (accumulate across relations).
  // C/D layout: VGPR j holds M = j + 8*half, N = l16.
  v8f c;
#pragma unroll
  for (int j = 0; j < 8; ++j)
    c[j] = out[(size_t)(rowBase + j + half * 8) * OUT_SIZE + colBase + l16];

  const float* aRow = agg + (size_t)(rowBase + l16) * IN_SIZE;
  const int koff = half * 2;  // A/B: lanes 0..15 hold K,K+1; lanes 16..31 hold K+2,K+3
#pragma unroll
  for (int k = 0; k < IN_SIZE; k += 4) {
    v2f a, b;
    const float2 av = *(const float2*)(aRow + k + koff);
    a[0] = av.x;
    a[1] = av.y;
    b[0] = sW[(k + koff) * OUT_SIZE + colBase + l16];
    b[1] = sW[(k + koff + 1) * OUT_SIZE + colBase + l16];
    // D = A(16x4) x B(4x16) + C, f32
    c = __builtin_amdgcn_wmma_f32_16x16x4_f32(
        /*neg_a=*/false, a, /*neg_b=*/false, b,
        /*c_mod=*/(short)0, c, /*reuse_a=*/false, /*reuse_b=*/false);
  }

#pragma unroll
  for (int j = 0; j < 8; ++j)
    out[(size_t)(rowBase + j + half * 8) * OUT_SIZE + colBase + l16] = c[j];
}

extern "C" void kernel_launch(void* const* d_in, const int* in_sizes, int n_in,
                              void* d_out, int out_size, void* d_ws, size_t ws_size,
                              hipStream_t stream) {
  const float* inp = (const float*)d_in[0];
  const int* src = (const int*)d_in[1];
  const int* dst = (const int*)d_in[2];
  const float* eval = (const float*)d_in[3];
  const float* weights = (const float*)d_in[4];
  const float* bias = (const float*)d_in[5];
  float* out = (float*)d_out;
  float* agg = (float*)d_ws;  // N * 64 floats (25.6 MB) reused per relation

  const int n_nodes = in_sizes[0] / IN_SIZE;
  const int n_rel = in_sizes[4] / (IN_SIZE * OUT_SIZE);
  const int n_edges = in_sizes[1] / n_rel;

  const int n4 = n_nodes * OUT_SIZE / 4;
  const int vb = (n4 + 255) / 256;
  rgcn_init_out<<<vb, 256, 0, stream>>>(bias, out, n4, n_rel);

  const int scatter_blocks = (n_edges * 32 + 255) / 256;  // one edge per wave
  const int num_tiles = (n_nodes / 16) * (OUT_SIZE / 16);
  const int gemm_blocks = (num_tiles + 7) / 8;

  for (int r = 0; r < n_rel; ++r) {
    rgcn_zero<<<vb, 256, 0, stream>>>((float4*)agg, n4);
    rgcn_scatter<<<scatter_blocks, 256, 0, stream>>>(
        inp, src + (size_t)r * n_edges, dst + (size_t)r * n_edges,
        eval + (size_t)r * n_edges, agg, n_edges);
    rgcn_gemm_wmma<<<gemm_blocks, 256, 0, stream>>>(
        agg, weights + (size_t)r * IN_SIZE * OUT_SIZE, out, num_tiles);
  }
}